// DrawRope_11647951307197
// MI455X (gfx1250) — compile-verified
//
#include <hip/hip_runtime.h>
#include <hip/hip_bf16.h>
#include <stdint.h>

#define IMAGE_H 200
#define IMAGE_W 200
#define FRAME_FLOATS (IMAGE_H * IMAGE_W * 3)   // 120000 floats per frame
#define FRAME_VEC4   (FRAME_FLOATS / 4)        // 30000 16B chunks per frame (exact)
#define N_STEPS      (IMAGE_H + IMAGE_W)       // 400, matches reference scan length
#define TPB          256                       // 8 wave32 waves per block

typedef float v4f __attribute__((ext_vector_type(4)));

__device__ __forceinline__ void plot_px(float* __restrict__ img, int r, int c,
                                        float v0, float v1, float v2) {
    // reference uses mode='drop' for OOB scatters
    if ((unsigned)r < (unsigned)IMAGE_H && (unsigned)c < (unsigned)IMAGE_W) {
        float* p = img + ((size_t)r * IMAGE_W + c) * 3;
        p[0] = v0; p[1] = v1; p[2] = v2;
    }
}

// Exact port of the reference's Bresenham scan body:
// emit pixel first, then test done, then conditional step (e2>=dy / e2<=dx).
__device__ void draw_line(float* __restrict__ img, int c0, int r0, int c1, int r1) {
    int dx = abs(c1 - c0);
    int dy = -abs(r1 - r0);
    int sc = (c0 < c1) ? 1 : -1;
    int sr = (r0 < r1) ? 1 : -1;
    int err = dx + dy;
    int c = c0, r = r0;
    #pragma unroll 1
    for (int i = 0; i < N_STEPS; ++i) {
        plot_px(img, r, c, 128.0f, 128.0f, 128.0f);
        if (c == c1 && r == r1) break;
        int e2 = 2 * err;
        if (e2 >= dy) { err += dy; c += sc; }
        if (e2 <= dx) { err += dx; r += sr; }
    }
}

__global__ void __launch_bounds__(TPB)
draw_rope_kernel(const float* __restrict__ x,          // [M, 6] = (x0,y0,x1,y1,x2,y2)
                 const float* __restrict__ resolution, // [2]
                 const float* __restrict__ origin,     // [2]
                 float* __restrict__ out) {            // [M, 200, 200, 3]
    const int m = blockIdx.x;
    float* __restrict__ img = out + (size_t)m * FRAME_FLOATS;

    // ---- Phase 1: zero the frame via the CDNA5 async LDS->memory mover.
    // Stage 4KB of zeros in LDS (one 16B chunk per lane), then stream the
    // whole 480KB frame with GLOBAL_STORE_ASYNC_FROM_LDS_B128 in GVS mode:
    // SGPR pair holds the uniform frame base, the per-lane VGPR holds the
    // raw chunk index, and SCALE_OFFSET multiplies it by the 16B data size
    // in hardware. The payload never passes through VGPRs; ASYNCcnt keeps
    // ~63 stores in flight per wave. 384MB of write-only output exceeds the
    // 192MB L2 -> th:TH_STORE_NT so the stream doesn't displace the cache.
    // Pure HBM-store-bandwidth phase: 384MB / 23.3TB/s ~= 16.5us floor.
    __shared__ alignas(16) v4f zstage[TPB];
    zstage[threadIdx.x] = (v4f){0.f, 0.f, 0.f, 0.f};
    // Async engine reads of LDS are NOT ordered against DS writes by any
    // shared counter: drain DScnt explicitly, then barrier.
    asm volatile("s_wait_dscnt 0x0" ::: "memory");
    __syncthreads();

    {
        // Low 32 bits of a flat shared pointer == LDS byte address
        // (shared aperture lives in addr[63:32]).
        unsigned lds_src = (unsigned)(uintptr_t)&zstage[threadIdx.x];
        #pragma unroll 1
        for (int i = threadIdx.x; i < FRAME_VEC4; i += TPB) {
            // mem_addr = SGPR64(img) + VGPR32(i) * 16  (coalesced 512B/wave)
            asm volatile(
                "global_store_async_from_lds_b128 %0, %1, %2 scale_offset th:TH_STORE_NT"
                :: "v"(i), "v"(lds_src), "s"(img)
                : "memory");
        }
        // ASYNCcnt is not covered by s_wait_storecnt / __threadfence: drain it
        // before the release fence so the Bresenham overwrites below are
        // ordered after the fill at the L2 coherence point.
        asm volatile("s_wait_asynccnt 0x0" ::: "memory");
    }

    __threadfence();
    __syncthreads();

    // ---- Phase 2: pixel indices for the 3 rope points (cheap; all threads).
    const float* p   = x + (size_t)m * 6;
    const float res0 = resolution[0], res1 = resolution[1];
    const float org0 = origin[0],     org1 = origin[1];
    int r[3], c[3];
    #pragma unroll
    for (int k = 0; k < 3; ++k) {
        // row from y with resolution[0]/origin[0]; col from x with [1]
        r[k] = (int)floorf(p[2 * k + 1] / res0 + org0);
        c[k] = (int)floorf(p[2 * k + 0] / res1 + org1);
    }

    // Two segments; both write the same value 128, so order between them is free.
    if (threadIdx.x == 0)      draw_line(img, c[0], r[0], c[1], r[1]);
    else if (threadIdx.x == 1) draw_line(img, c[1], r[1], c[2], r[2]);

    __threadfence();
    __syncthreads();

    // ---- Phase 3: endpoints, in reference scatter order (last write wins).
    if (threadIdx.x == 0) {
        plot_px(img, r[0], c[0], 255.0f,   0.0f, 0.0f);  // red
        plot_px(img, r[1], c[1], 255.0f,   0.0f, 0.0f);  // red
        plot_px(img, r[2], c[2],   0.0f, 255.0f, 0.0f);  // green (last)
    }
}

extern "C" void kernel_launch(void* const* d_in, const int* in_sizes, int n_in,
                              void* d_out, int out_size, void* d_ws, size_t ws_size,
                              hipStream_t stream) {
    const float* x          = (const float*)d_in[0]; // [B*T*6]
    const float* resolution = (const float*)d_in[1]; // [2]
    const float* origin     = (const float*)d_in[2]; // [2]
    float* out              = (float*)d_out;

    const int frames = in_sizes[0] / 6;              // B*T = 800
    if (frames <= 0) return;

    draw_rope_kernel<<<dim3(frames), dim3(TPB), 0, stream>>>(x, resolution, origin, out);
}